// AnomalousTokenPerception_90417651516041
// MI455X (gfx1250) — compile-verified
//
#include <hip/hip_runtime.h>
#include <hip/hip_bf16.h>

typedef float v2f __attribute__((ext_vector_type(2)));
typedef float v4f __attribute__((ext_vector_type(4)));
typedef float v8f __attribute__((ext_vector_type(8)));

#define N_INST 256
#define HW (1024 * 1024)          // elements per instance
#define BPI 64                    // blocks per instance
#define CHUNK (HW / BPI)          // 16384 floats per block
#define T1 256                    // threads, pass-1 block

// ---------------------------------------------------------------------------
// Pass 1: streaming partial reduction. Each block consumes one 64 KB chunk of
// one instance with non-temporal float4 loads and writes (sum, sumsq, count)
// to a fixed workspace slot. Deterministic: no float atomics, fixed slots.
// ---------------------------------------------------------------------------
__global__ void __launch_bounds__(T1)
k_partials(const float* __restrict__ x, float* __restrict__ part) {
    const int inst = blockIdx.x >> 6;       // /BPI
    const int blk  = blockIdx.x & (BPI - 1);
    const size_t base = (size_t)inst * HW + (size_t)blk * CHUNK;
    const v4f* __restrict__ p = (const v4f*)(x + base);
    const int tid = threadIdx.x;

    float sum = 0.0f, sq = 0.0f;
    int cnt = 0;
#pragma unroll
    for (int it = 0; it < CHUNK / (4 * T1); ++it) {   // 16 iters
        v4f v = __builtin_nontemporal_load(&p[it * T1 + tid]);
#pragma unroll
        for (int c = 0; c < 4; ++c) {
            float xv = v[c];
            float pr = 1.0f / (1.0f + __expf(-xv));   // sigmoid
            sum += pr;
            sq  += pr * pr;
            cnt += (xv > 0.0f) ? 1 : 0;               // sigmoid(x)>0.5 <=> x>0
        }
    }
    float fc = (float)cnt;                            // <= 64 per thread: exact

    // wave32 reduction
#pragma unroll
    for (int m = 16; m >= 1; m >>= 1) {
        sum += __shfl_xor(sum, m);
        sq  += __shfl_xor(sq, m);
        fc  += __shfl_xor(fc, m);
    }

    __shared__ float s0[T1 / 32], s1[T1 / 32], s2[T1 / 32];
    const int wave = tid >> 5;
    if ((tid & 31) == 0) { s0[wave] = sum; s1[wave] = sq; s2[wave] = fc; }
    __syncthreads();
    if (tid == 0) {
        float a = 0.0f, b = 0.0f, c = 0.0f;
#pragma unroll
        for (int i = 0; i < T1 / 32; ++i) { a += s0[i]; b += s1[i]; c += s2[i]; }
        float* o = part + (size_t)blockIdx.x * 4;
        o[0] = a; o[1] = b; o[2] = c; o[3] = 0.0f;
    }
}

// ---------------------------------------------------------------------------
// Pass 2: per-instance final reduction (in double), build normalized
// embedding E[inst] = normalize([mean, std(ddof=1), count]), K-padded to 4.
// ---------------------------------------------------------------------------
__global__ void __launch_bounds__(64)
k_emb(const float* __restrict__ part, float* __restrict__ E) {
    const int inst = blockIdx.x;
    const int tid  = threadIdx.x;
    const float* pp = part + ((size_t)inst * BPI + tid) * 4;

    __shared__ double d0[BPI], d1[BPI], d2[BPI];
    d0[tid] = (double)pp[0];
    d1[tid] = (double)pp[1];
    d2[tid] = (double)pp[2];
    __syncthreads();
#pragma unroll
    for (int s = BPI / 2; s > 0; s >>= 1) {
        if (tid < s) { d0[tid] += d0[tid + s]; d1[tid] += d1[tid + s]; d2[tid] += d2[tid + s]; }
        __syncthreads();
    }
    if (tid == 0) {
        const double n   = (double)HW;
        const double sum = d0[0], sq = d1[0], cnt = d2[0];
        const double mean = sum / n;
        double var = (sq - sum * sum / n) / (n - 1.0);   // unbiased (ddof=1)
        var = var > 0.0 ? var : 0.0;
        const double sd = sqrt(var);
        double nrm = sqrt(mean * mean + sd * sd + cnt * cnt);
        nrm = nrm > 1e-12 ? nrm : 1e-12;
        float* e = E + inst * 4;
        e[0] = (float)(mean / nrm);
        e[1] = (float)(sd   / nrm);
        e[2] = (float)(cnt  / nrm);
        e[3] = 0.0f;                                     // K pad
    }
}

// ---------------------------------------------------------------------------
// Pass 3: sim = E * E^T via V_WMMA_F32_16X16X4_F32. One block, 16 waves.
// Wave m owns M-tile m and chains 16 WMMAs (C<-D) across all N-tiles, so the
// accumulator holds per-row partial sums of the whole sim row block.
// A 16x4 f32 layout: lanes 0-15 row M=lane {K0,K1}; lanes 16-31 same rows
// {K2,K3}. B 4x16: lane j holds column j (= E row of the N tile), same K
// split. C/D: VGPR r -> row r (lanes 0-15) and row r+8 (lanes 16-31).
// ---------------------------------------------------------------------------
__global__ void __launch_bounds__(512)
k_sim(const float* __restrict__ E, float* __restrict__ out) {
    const int tid  = threadIdx.x;
    const int wave = tid >> 5;
    const int lane = tid & 31;
    const int l15  = lane & 15;
    const int kOff = (lane >> 4) * 2;      // lanes 0-15: K=0,1 ; lanes 16-31: K=2,3

    const int rowA = wave * 16 + l15;
    v2f a;
    a.x = E[rowA * 4 + kOff];
    a.y = E[rowA * 4 + kOff + 1];

    v8f acc = {};                          // C starts at zero
    for (int n = 0; n < 16; ++n) {
        const int rowB = n * 16 + l15;
        v2f b;
        b.x = E[rowB * 4 + kOff];
        b.y = E[rowB * 4 + kOff + 1];
        acc = __builtin_amdgcn_wmma_f32_16x16x4_f32(
            /*neg_a=*/false, a, /*neg_b=*/false, b,
            /*c_mod=*/(short)0, acc, /*reuse_a=*/false, /*reuse_b=*/false);
    }

    __shared__ float rowsum[N_INST];
#pragma unroll
    for (int r = 0; r < 8; ++r) {
        float v = acc[r];
        // reduce within each 16-lane half (row sums over N)
        v += __shfl_xor(v, 1);
        v += __shfl_xor(v, 2);
        v += __shfl_xor(v, 4);
        v += __shfl_xor(v, 8);
        if (lane == 0)  rowsum[wave * 16 + r]     = v;  // rows 0..7 of tile
        if (lane == 16) rowsum[wave * 16 + r + 8] = v;  // rows 8..15 of tile
    }
    __syncthreads();

    __shared__ float red[N_INST];
    if (tid < N_INST) red[tid] = rowsum[tid];
    __syncthreads();
#pragma unroll
    for (int s = N_INST / 2; s > 0; s >>= 1) {
        if (tid < s) red[tid] += red[tid + s];
        __syncthreads();
    }
    if (tid < N_INST) {
        const float avg  = rowsum[tid] * (1.0f / (float)N_INST);
        const float mean = red[0] * (1.0f / ((float)N_INST * (float)N_INST));
        const float z    = avg - mean;
        out[tid] = 1.0f / (1.0f + __expf(-z));
    }
}

// ---------------------------------------------------------------------------
extern "C" void kernel_launch(void* const* d_in, const int* in_sizes, int n_in,
                              void* d_out, int out_size, void* d_ws, size_t ws_size,
                              hipStream_t stream) {
    (void)in_sizes; (void)n_in; (void)out_size; (void)ws_size;
    const float* x = (const float*)d_in[0];       // (256, 1024, 1024) fp32
    float* out = (float*)d_out;                   // (256,) fp32

    float* part = (float*)d_ws;                   // 16384 * 4 floats = 256 KB
    float* E    = part + (size_t)N_INST * BPI * 4;// 256 * 4 floats

    k_partials<<<dim3(N_INST * BPI), dim3(T1), 0, stream>>>(x, part);
    k_emb<<<dim3(N_INST), dim3(BPI), 0, stream>>>(part, E);
    k_sim<<<dim3(1), dim3(512), 0, stream>>>(E, out);
}